// S_MSE_43181601194361
// MI455X (gfx1250) — compile-verified
//
#include <hip/hip_runtime.h>

// ---------------------------------------------------------------------------
// S_MSE on MI455X (gfx1250):  per-segment  (2*n*Σ||x||² - 2*||Σx||²)/(n(n-1))/D,
// nanmean over segments.  Memory-bound streaming pass (512 MB @ 23.3 TB/s ≈ 22us).
// Segment sums computed as Ones(16x4) × Xtile(4x16) via V_WMMA_F32_16X16X4_F32,
// accumulated in the f32 WMMA accumulator; squares via a second WMMA on b∘b.
// ---------------------------------------------------------------------------

typedef __attribute__((ext_vector_type(2))) float v2f;
typedef __attribute__((ext_vector_type(8))) float v8f;

#define WAVE            32
#define WAVES_PER_BLOCK 8
#define TOK_CHUNK       256   // tokens per wave
#define FEAT_SLICE      16    // features per wave (WMMA N)

__global__ void __launch_bounds__(WAVE * WAVES_PER_BLOCK)
smse_partial_kernel(const float* __restrict__ x,
                    const int*   __restrict__ segid,
                    float*       __restrict__ ws_sx,   // [B, D] feature sums
                    float*       __restrict__ ws_s2,   // [B]    sums of squares
                    int Dd, int featSlices, long totalWaves)
{
    const int  lane = threadIdx.x & (WAVE - 1);
    const long wid  = (long)blockIdx.x * WAVES_PER_BLOCK + (threadIdx.x >> 5);
    if (wid >= totalWaves) return;   // uniform per wave: EXEC stays all-ones

    const int  fslice = (int)(wid % featSlices);
    const long chunk  = wid / featSlices;
    const long token0 = chunk * TOK_CHUNK;
    const int  f0     = fslice * FEAT_SLICE;
    const int  seg    = segid[token0];   // chunk never crosses a segment here

    // B-matrix tile (4 tokens x 16 features): each lane supplies 2 dwords.
    //   half = lane>>4 selects token pair {t,t+1} vs {t+2,t+3}
    //   col  = lane&15 selects the feature column (N)
    const int half = lane >> 4;
    const int col  = lane & 15;

    v8f c  = {};            // accumulates per-feature sums
    v8f c2 = {};            // accumulates per-feature sums of squares
    v2f ones; ones[0] = 1.0f; ones[1] = 1.0f;   // A = 16x4 all-ones

    const float* base = x + (token0 + (long)(half * 2)) * (long)Dd + f0 + col;
    #pragma unroll 4
    for (int it = 0; it < TOK_CHUNK / 4; ++it) {
        const float* p = base + (long)it * 4 * Dd;
        v2f b, bs;
        b[0]  = __builtin_nontemporal_load(p);        // row t + 2*half
        b[1]  = __builtin_nontemporal_load(p + Dd);   // row t + 2*half + 1
        bs[0] = b[0] * b[0];
        bs[1] = b[1] * b[1];
        // D = Ones(16x4) * B(4x16) + C : every row of D = column sums of tile
        c  = __builtin_amdgcn_wmma_f32_16x16x4_f32(false, ones, false, b,
                                                   (short)0, c,  false, false);
        c2 = __builtin_amdgcn_wmma_f32_16x16x4_f32(false, ones, false, bs,
                                                   (short)0, c2, false, false);
    }

    // C/D layout: VGPR0, lanes 0..15 -> M=0,N=lane (lanes 16..31 hold M=8, a
    // duplicate row since all rows equal the column sums). Use lanes 0..15.
    if (lane < 16) {
        atomicAdd(&ws_sx[(long)seg * Dd + f0 + lane], c[0]);
    }

    // Reduce the 16 per-feature square-sums within each 16-lane group.
    float sq = c2[0];
    sq += __shfl_xor(sq, 1, WAVE);
    sq += __shfl_xor(sq, 2, WAVE);
    sq += __shfl_xor(sq, 4, WAVE);
    sq += __shfl_xor(sq, 8, WAVE);
    if (lane == 0) {
        atomicAdd(&ws_s2[seg], sq);
    }
}

__global__ void __launch_bounds__(256)
smse_finalize_kernel(const float* __restrict__ ws_sx,
                     const float* __restrict__ ws_s2,
                     const int*   __restrict__ lengths,
                     float*       __restrict__ out,
                     int Bseg, int Dd)
{
    __shared__ float red[256];
    __shared__ float acc_sum;
    __shared__ float acc_cnt;
    const int tid = threadIdx.x;
    if (tid == 0) { acc_sum = 0.0f; acc_cnt = 0.0f; }
    __syncthreads();

    for (int s = 0; s < Bseg; ++s) {
        float p = 0.0f;
        for (int d = tid; d < Dd; d += 256) {
            float v = ws_sx[(long)s * Dd + d];
            p += v * v;                              // ||sx||^2 partial
        }
        red[tid] = p;
        __syncthreads();
        for (int off = 128; off > 0; off >>= 1) {
            if (tid < off) red[tid] += red[tid + off];
            __syncthreads();
        }
        if (tid == 0) {
            float n     = (float)lengths[s];
            float total = 2.0f * n * ws_s2[s] - 2.0f * red[0];
            float per   = total / (n * (n - 1.0f)) / (float)Dd;
            if (!(per != per)) {                     // nanmean: skip NaNs
                acc_sum += per;
                acc_cnt += 1.0f;
            }
        }
        __syncthreads();
    }
    if (tid == 0) {
        out[0] = (acc_cnt > 0.0f) ? (acc_sum / acc_cnt)
                                  : __builtin_nanf("");
    }
}

extern "C" void kernel_launch(void* const* d_in, const int* in_sizes, int n_in,
                              void* d_out, int out_size, void* d_ws, size_t ws_size,
                              hipStream_t stream)
{
    const float* inputs  = (const float*)d_in[0];   // [T, D] f32
    const int*   segids  = (const int*)  d_in[1];   // [T]
    const int*   lengths = (const int*)  d_in[2];   // [B]

    const int T    = in_sizes[1];
    const int Bseg = in_sizes[2];
    const int Dd   = in_sizes[0] / T;               // 512

    float* ws_sx = (float*)d_ws;                    // [B, D]
    float* ws_s2 = ws_sx + (size_t)Bseg * Dd;       // [B]
    hipMemsetAsync(d_ws, 0, ((size_t)Bseg * Dd + Bseg) * sizeof(float), stream);

    const int  featSlices = Dd / FEAT_SLICE;        // 32
    const long chunks     = (long)T / TOK_CHUNK;    // 1024
    const long totalWaves = chunks * featSlices;    // 32768
    const long grid       = (totalWaves + WAVES_PER_BLOCK - 1) / WAVES_PER_BLOCK;

    smse_partial_kernel<<<(int)grid, WAVE * WAVES_PER_BLOCK, 0, stream>>>(
        inputs, segids, ws_sx, ws_s2, Dd, featSlices, totalWaves);

    smse_finalize_kernel<<<1, 256, 0, stream>>>(
        ws_sx, ws_s2, lengths, (float*)d_out, Bseg, Dd);
}